// BatchMultiHeadGraphAttention_49735721288093
// MI455X (gfx1250) — compile-verified
//
#include <hip/hip_runtime.h>

typedef __attribute__((ext_vector_type(2))) float v2f;
typedef __attribute__((ext_vector_type(8))) float v8f;

#define BS   8
#define NH   4
#define NN   2048
#define FIN  64
#define FOUT 32

static __device__ __forceinline__ float lrelu(float x) {
    return x > 0.0f ? x : 0.2f * x;
}

// ---------------------------------------------------------------------------
// Kernel 1: h_prime[b,h,i,o] = sum_f h[b,i,f] * w[h,f,o]
// One wave per (b,h, 16-row tile). WMMA f32 16x16x4, K=64 -> 16 k-steps,
// two 16-wide N tiles (FOUT=32).
// ---------------------------------------------------------------------------
__global__ __launch_bounds__(256) void k_hprime(const float* __restrict__ h,
                                                const float* __restrict__ w,
                                                float* __restrict__ hp) {
    const int lane  = threadIdx.x & 31;
    const int wave  = blockIdx.x * 8 + (threadIdx.x >> 5);
    const int mt    = wave & 127;       // 128 M tiles
    const int bh    = wave >> 7;        // b*NH + head
    const int head  = bh & (NH - 1);
    const int b     = bh >> 2;
    const int l15   = lane & 15;
    const int khalf = (lane >> 4) << 1; // 0 or 2

    const float* hrow = h + ((size_t)b * NN + (size_t)mt * 16 + l15) * FIN;
    const float* wmat = w + (size_t)head * FIN * FOUT;

    v8f acc0 = {0.f,0.f,0.f,0.f,0.f,0.f,0.f,0.f};
    v8f acc1 = {0.f,0.f,0.f,0.f,0.f,0.f,0.f,0.f};

    #pragma unroll
    for (int k4 = 0; k4 < FIN / 4; ++k4) {
        const int kb = k4 * 4 + khalf;
        v2f a;  a.x  = hrow[kb];                     a.y  = hrow[kb + 1];
        v2f b0; b0.x = wmat[kb * FOUT + l15];        b0.y = wmat[(kb + 1) * FOUT + l15];
        v2f b1; b1.x = wmat[kb * FOUT + l15 + 16];   b1.y = wmat[(kb + 1) * FOUT + l15 + 16];
        acc0 = __builtin_amdgcn_wmma_f32_16x16x4_f32(false, a, false, b0, (short)0, acc0, false, false);
        acc1 = __builtin_amdgcn_wmma_f32_16x16x4_f32(false, a, false, b1, (short)0, acc1, false, false);
    }

    float* hpb = hp + ((size_t)bh * NN + (size_t)mt * 16) * FOUT;
    const int mofs = (lane >> 4) << 3;  // 0 or 8
    #pragma unroll
    for (int vi = 0; vi < 8; ++vi) {
        const int m = vi + mofs;
        hpb[m * FOUT + l15]      = acc0[vi];
        hpb[m * FOUT + l15 + 16] = acc1[vi];
    }
}

// ---------------------------------------------------------------------------
// Kernel 2: per-row score dots  s = hp_row . a_src[head],  d = hp_row . a_dst[head]
// ---------------------------------------------------------------------------
__global__ __launch_bounds__(256) void k_scores(const float* __restrict__ hp,
                                                const float* __restrict__ a_src,
                                                const float* __restrict__ a_dst,
                                                float* __restrict__ ssrc,
                                                float* __restrict__ sdst) {
    const int t = blockIdx.x * blockDim.x + threadIdx.x;   // [0, BS*NH*NN)
    const int head = (t / NN) & (NH - 1);
    const float4* row = (const float4*)(hp + (size_t)t * FOUT);
    const float4* as  = (const float4*)(a_src + head * FOUT);
    const float4* ad  = (const float4*)(a_dst + head * FOUT);
    float s = 0.f, d = 0.f;
    #pragma unroll
    for (int q = 0; q < FOUT / 4; ++q) {
        const float4 v = row[q], x = as[q], y = ad[q];
        s += v.x * x.x + v.y * x.y + v.z * x.z + v.w * x.w;
        d += v.x * y.x + v.y * y.y + v.z * y.z + v.w * y.w;
    }
    ssrc[t] = s;
    sdst[t] = d;
}

// ---------------------------------------------------------------------------
// Kernel 3: maxd[bh] = max_j d[bh, j]   (lrelu monotone -> row max = lrelu(s+maxd))
// ---------------------------------------------------------------------------
__global__ __launch_bounds__(256) void k_maxd(const float* __restrict__ sdst,
                                              float* __restrict__ maxd) {
    __shared__ float red[256];
    const float* dvec = sdst + (size_t)blockIdx.x * NN;
    float m = -3.4e38f;
    for (int j = threadIdx.x; j < NN; j += 256) m = fmaxf(m, dvec[j]);
    red[threadIdx.x] = m;
    __syncthreads();
    for (int s = 128; s > 0; s >>= 1) {
        if (threadIdx.x < s) red[threadIdx.x] = fmaxf(red[threadIdx.x], red[threadIdx.x + s]);
        __syncthreads();
    }
    if (threadIdx.x == 0) maxd[blockIdx.x] = red[0];
}

// ---------------------------------------------------------------------------
// Kernel 4 (fused): per 16-row query block:
//   Z-pass over 2048 keys (d in LDS), then per 64-key chunk:
//     compute normalized P tile into LDS -> NT-store attn (coalesced rows)
//     -> WMMA accumulate  O += P @ h_prime   (f32 16x16x4)
// Block = 8 waves, each wave owns one 16-row block; d[2048] shared in LDS.
// ---------------------------------------------------------------------------
__global__ __launch_bounds__(256) void k_attn(const float* __restrict__ hp,
                                              const float* __restrict__ ssrc,
                                              const float* __restrict__ sdst,
                                              const float* __restrict__ maxd,
                                              const float* __restrict__ bias,
                                              float* __restrict__ out,
                                              float* __restrict__ attn) {
    __shared__ float sd[NN];             // 8 KB
    __shared__ float stage[8][16][68];   // 34 KB, padded stride 68 (bank-conflict free)
    __shared__ float s_s[8][16];
    __shared__ float s_m[8][16];
    __shared__ float s_iz[8][16];

    const int bh   = blockIdx.x >> 4;    // 32 (b,h) pairs
    const int mtg  = blockIdx.x & 15;    // 16 groups of 8 row-tiles
    const int w    = threadIdx.x >> 5;
    const int lane = threadIdx.x & 31;
    const int l15  = lane & 15;
    const int half = lane >> 4;

    for (int j = threadIdx.x; j < NN; j += 256) sd[j] = sdst[(size_t)bh * NN + j];

    const float md = maxd[bh];
    const int   i0 = (mtg * 8 + w) * 16;

    if (lane < 16) {
        const float s = ssrc[(size_t)bh * NN + i0 + lane];
        s_s[w][lane] = s;
        s_m[w][lane] = lrelu(s + md);
    }
    __syncthreads();

    // Z pass: lane covers row l15, keys j = half, half+2, ...
    {
        const float sr = s_s[w][l15];
        const float mr = s_m[w][l15];
        float z = 0.f;
        for (int j = half; j < NN; j += 2)
            z += __expf(lrelu(sr + sd[j]) - mr);
        z += __shfl_xor(z, 16);
        if (lane < 16) s_iz[w][lane] = 1.0f / z;
    }
    __syncthreads();

    const int khalf = half << 1;
    v8f acc0 = {0.f,0.f,0.f,0.f,0.f,0.f,0.f,0.f};
    v8f acc1 = {0.f,0.f,0.f,0.f,0.f,0.f,0.f,0.f};

    const float* vbase = hp + (size_t)bh * NN * FOUT;
    float* arow = attn + ((size_t)bh * NN + i0) * NN;

    for (int c = 0; c < NN / 64; ++c) {
        __syncthreads();   // guard stage reuse across iterations
        const int j0 = c * 64;
        // phase A: normalized probabilities for this 16x64 tile into LDS
        #pragma unroll
        for (int r = 0; r < 16; ++r) {
            const float sr = s_s[w][r], mr = s_m[w][r], iz = s_iz[w][r];
            stage[w][r][lane]      = __expf(lrelu(sr + sd[j0 + lane])      - mr) * iz;
            stage[w][r][lane + 32] = __expf(lrelu(sr + sd[j0 + lane + 32]) - mr) * iz;
        }
        __syncthreads();
        // phase B: stream attn to HBM, non-temporal, 256B-contiguous per row
        #pragma unroll
        for (int r = 0; r < 16; ++r) {
            v2f v; v.x = stage[w][r][lane * 2]; v.y = stage[w][r][lane * 2 + 1];
            __builtin_nontemporal_store(v, (v2f*)(arow + (size_t)r * NN + j0 + lane * 2));
        }
        // phase C: O += P(16x64) @ V(64x32) via 16 k-steps x 2 N-tiles
        #pragma unroll
        for (int k4 = 0; k4 < 16; ++k4) {
            const int kb = k4 * 4 + khalf;
            v2f a;  a.x  = stage[w][l15][kb];      a.y  = stage[w][l15][kb + 1];
            const float* vrow = vbase + (size_t)(j0 + kb) * FOUT;
            v2f b0; b0.x = vrow[l15];              b0.y = vrow[FOUT + l15];
            v2f b1; b1.x = vrow[l15 + 16];         b1.y = vrow[FOUT + l15 + 16];
            acc0 = __builtin_amdgcn_wmma_f32_16x16x4_f32(false, a, false, b0, (short)0, acc0, false, false);
            acc1 = __builtin_amdgcn_wmma_f32_16x16x4_f32(false, a, false, b1, (short)0, acc1, false, false);
        }
    }

    const float bia0 = bias[l15], bia1 = bias[l15 + 16];
    float* ob = out + ((size_t)bh * NN + i0) * FOUT;
    const int mofs = half << 3;
    #pragma unroll
    for (int vi = 0; vi < 8; ++vi) {
        const int m = vi + mofs;
        ob[m * FOUT + l15]      = acc0[vi] + bia0;
        ob[m * FOUT + l15 + 16] = acc1[vi] + bia1;
    }
}

// ---------------------------------------------------------------------------
extern "C" void kernel_launch(void* const* d_in, const int* in_sizes, int n_in,
                              void* d_out, int out_size, void* d_ws, size_t ws_size,
                              hipStream_t stream) {
    const float* h     = (const float*)d_in[0];   // [8,2048,64]
    const float* w     = (const float*)d_in[1];   // [4,64,32]
    const float* a_src = (const float*)d_in[2];   // [4,32,1]
    const float* a_dst = (const float*)d_in[3];   // [4,32,1]
    const float* bias  = (const float*)d_in[4];   // [32]

    float* out  = (float*)d_out;                          // [8,4,2048,32]
    float* attn = out + (size_t)BS * NH * NN * FOUT;      // [8,4,2048,2048]

    float* ws   = (float*)d_ws;
    float* hp   = ws;                                     // 2,097,152 floats
    float* ssrc = hp + (size_t)BS * NH * NN * FOUT;       // 65,536
    float* sdst = ssrc + (size_t)BS * NH * NN;            // 65,536
    float* maxd = sdst + (size_t)BS * NH * NN;            // 32

    k_hprime<<<512, 256, 0, stream>>>(h, w, hp);
    k_scores<<<(BS * NH * NN) / 256, 256, 0, stream>>>(hp, a_src, a_dst, ssrc, sdst);
    k_maxd  <<<BS * NH, 256, 0, stream>>>(sdst, maxd);
    k_attn  <<<512, 256, 0, stream>>>(hp, ssrc, sdst, maxd, bias, out, attn);
}